// LinearKalmanFilter_36704790512128
// MI455X (gfx1250) — compile-verified
//
#include <hip/hip_runtime.h>
#include <hip/hip_bf16.h>

#define NXX 64
#define NYY 32
#define NUU 16
#define NDD 8
#define T_STEPS 2048
#define B_TOT 256

typedef float v2f __attribute__((ext_vector_type(2)));
typedef float v8f __attribute__((ext_vector_type(8)));

__device__ __forceinline__ v8f zero8() {
    v8f c;
#pragma unroll
    for (int r = 0; r < 8; ++r) c[r] = 0.0f;
    return c;
}

// ---- WMMA f32 16x16x4 fragment helpers (ISA 7.12.2 32-bit layouts) ----
// A (16x4, MxK): lanes 0-15 hold row M=lane, K=k0..k0+1; lanes 16-31 hold K=k0+2..k0+3
__device__ __forceinline__ v2f fragA(const float* X, int ld, int m0, int k0, int lane) {
    int h = lane >> 4, m = m0 + (lane & 15), k = k0 + 2 * h;
    v2f a; a.x = X[m * ld + k]; a.y = X[m * ld + k + 1]; return a;
}
// A transposed: A[m][k] = X[k][m]
__device__ __forceinline__ v2f fragAT(const float* X, int ld, int m0, int k0, int lane) {
    int h = lane >> 4, m = m0 + (lane & 15), k = k0 + 2 * h;
    v2f a; a.x = X[k * ld + m]; a.y = X[(k + 1) * ld + m]; return a;
}
// B (4x16, KxN): lanes 0-15 hold rows K=k0,k0+1 at N=lane; lanes 16-31 rows K=k0+2,k0+3
__device__ __forceinline__ v2f fragB(const float* X, int ld, int k0, int n0, int lane) {
    int h = lane >> 4, n = n0 + (lane & 15), k = k0 + 2 * h;
    v2f b; b.x = X[k * ld + n]; b.y = X[(k + 1) * ld + n]; return b;
}
// B transposed: B[k][n] = X[n][k]
__device__ __forceinline__ v2f fragBT(const float* X, int ld, int k0, int n0, int lane) {
    int h = lane >> 4, n = n0 + (lane & 15), k = k0 + 2 * h;
    v2f b; b.x = X[n * ld + k]; b.y = X[n * ld + k + 1]; return b;
}
// C/D (16x16): VGPR r, lanes 0-15 -> [M=r][N=lane], lanes 16-31 -> [M=r+8][N=lane-16]
__device__ __forceinline__ void storeD(float* X, int ld, int m0, int n0, v8f d, int lane,
                                       float sgn = 1.0f) {
    int h = lane >> 4, n = n0 + (lane & 15);
#pragma unroll
    for (int r = 0; r < 8; ++r) X[(m0 + r + 8 * h) * ld + n] = sgn * d[r];
}
__device__ __forceinline__ v8f loadC(const float* X, int ld, int m0, int n0, int lane) {
    int h = lane >> 4, n = n0 + (lane & 15);
    v8f c;
#pragma unroll
    for (int r = 0; r < 8; ++r) c[r] = X[(m0 + r + 8 * h) * ld + n];
    return c;
}
__device__ __forceinline__ v8f identC(int m0, int n0, int lane) {
    int h = lane >> 4, n = n0 + (lane & 15);
    v8f c;
#pragma unroll
    for (int r = 0; r < 8; ++r) c[r] = ((m0 + r + 8 * h) == n) ? 1.0f : 0.0f;
    return c;
}

template <bool TA, bool TB>
__device__ __forceinline__ v8f mmacc(const float* A, int lda, int m0,
                                     const float* B, int ldb, int n0,
                                     int K, v8f c, int lane) {
#pragma unroll
    for (int k = 0; k < K; k += 4) {
        v2f a = TA ? fragAT(A, lda, m0, k, lane) : fragA(A, lda, m0, k, lane);
        v2f b = TB ? fragBT(B, ldb, k, n0, lane) : fragB(B, ldb, k, n0, lane);
        c = __builtin_amdgcn_wmma_f32_16x16x4_f32(false, a, false, b, (short)0, c,
                                                  false, false);
    }
    return c;
}

// ---------------------------------------------------------------------------
// Kernel 1: sequential covariance/gain recursion (batch-independent).
// Emits G_t = L^T (32x64 row-major), where L is the gain *carried into* step t
// (L_0 = 0, matching the reference's use of the previous step's gain).
// Single workgroup, 8 waves, everything in LDS, f32 WMMA throughout.
// ---------------------------------------------------------------------------
__global__ __launch_bounds__(256) void kf_gain_seq(const float* __restrict__ Wfx,
                                                   const float* __restrict__ Wfy,
                                                   float* __restrict__ G) {
    __shared__ float sWfx[NXX * NXX];   // 16 KB
    __shared__ float sWfy[NXX * NYY];   // 8 KB
    __shared__ float sP[NXX * NXX];     // 16 KB
    __shared__ float sM1[NXX * NXX];    // 16 KB  (Wfx*P, later -(Wfy^T*P))
    __shared__ float sM2[NXX * NYY];    // 8 KB   (P*Wfy)
    __shared__ float sSa[NYY * 64];     // 8 KB   (augmented [S | I] for GJ)
    __shared__ float sL[NXX * NYY];     // 8 KB

    const int tid = threadIdx.x, lane = tid & 31, w = tid >> 5;

    for (int i = tid; i < NXX * NXX; i += 256) {
        sWfx[i] = Wfx[i];
        sP[i] = ((i >> 6) == (i & 63)) ? 1.0f : 0.0f;   // P0 = I
    }
    for (int i = tid; i < NXX * NYY; i += 256) {
        sWfy[i] = Wfy[i];
        sL[i] = 0.0f;                                    // L0 = 0
    }
    __syncthreads();

    for (int t = 0; t < T_STEPS; ++t) {
        // emit G_t = sL^T  (row-major [32 x 64])
        float* Gt = G + (size_t)t * (NYY * NXX);
        for (int i = tid; i < NYY * NXX; i += 256) {
            int j = i >> 6, col = i & 63;
            Gt[i] = sL[col * NYY + j];
        }
        __syncthreads();

        // 1) M1 = Wfx * P
        for (int tt = w; tt < 16; tt += 8) {
            int mi = (tt >> 2) * 16, ni = (tt & 3) * 16;
            v8f c = zero8();
            c = mmacc<false, false>(sWfx, NXX, mi, sP, NXX, ni, NXX, c, lane);
            storeD(sM1, NXX, mi, ni, c, lane);
        }
        __syncthreads();
        // 2) P = M1 * Wfx^T + I        (Q = I)
        for (int tt = w; tt < 16; tt += 8) {
            int mi = (tt >> 2) * 16, ni = (tt & 3) * 16;
            v8f c = identC(mi, ni, lane);
            c = mmacc<false, true>(sM1, NXX, mi, sWfx, NXX, ni, NXX, c, lane);
            storeD(sP, NXX, mi, ni, c, lane);
        }
        __syncthreads();
        // 3) M2 = P * Wfy   (64x32, 8 tiles)
        {
            int mi = (w >> 1) * 16, ni = (w & 1) * 16;
            v8f c = zero8();
            c = mmacc<false, false>(sP, NXX, mi, sWfy, NYY, ni, NXX, c, lane);
            storeD(sM2, NYY, mi, ni, c, lane);
        }
        __syncthreads();
        // 4) Saug left = R + Wfy^T * M2 (R = I); right = I
        if (w < 4) {
            int mi = (w >> 1) * 16, ni = (w & 1) * 16;
            v8f c = identC(mi, ni, lane);
            c = mmacc<true, false>(sWfy, NYY, mi, sM2, NYY, ni, NXX, c, lane);
            storeD(sSa, 64, mi, ni, c, lane);
        } else {
            int r0 = (w - 4) * 8;
#pragma unroll
            for (int rr = 0; rr < 8; ++rr) {
                int r = r0 + rr;
                sSa[r * 64 + 32 + lane] = (lane == r) ? 1.0f : 0.0f;
            }
        }
        __syncthreads();
        // 5) Gauss-Jordan invert S in-place (no pivoting; S is SPD-ish, R=I)
        {
            const int r = tid >> 3, cs = (tid & 7) * 8;
            for (int p = 0; p < 32; ++p) {
                float piv = sSa[p * 64 + p];
                float f = sSa[r * 64 + p];
                float pr[8];
#pragma unroll
                for (int j = 0; j < 8; ++j) pr[j] = sSa[p * 64 + cs + j];
                __syncthreads();
                float ip = 1.0f / piv;
                if (r == p) {
#pragma unroll
                    for (int j = 0; j < 8; ++j) sSa[p * 64 + cs + j] = pr[j] * ip;
                } else {
#pragma unroll
                    for (int j = 0; j < 8; ++j) sSa[r * 64 + cs + j] -= f * pr[j] * ip;
                }
                __syncthreads();
            }
        }
        // 6) L = M2 * invS   (invS = sSa cols 32..63)
        {
            int mi = (w >> 1) * 16, ni = (w & 1) * 16;
            v8f c = zero8();
            c = mmacc<false, false>(sM2, NYY, mi, sSa + 32, 64, ni, NYY, c, lane);
            storeD(sL, NYY, mi, ni, c, lane);
        }
        __syncthreads();
        // 7) M1[0:32,:] = -(Wfy^T * P)   (negate at store; f32 WMMA has C-neg only)
        {
            int mi = (w >> 2) * 16, ni = (w & 3) * 16;
            v8f c = zero8();
            c = mmacc<true, false>(sWfy, NYY, mi, sP, NXX, ni, NXX, c, lane);
            storeD(sM1, NXX, mi, ni, c, lane, -1.0f);
        }
        __syncthreads();
        // 8) P = I + L * (-Wfy^T P)
        for (int tt = w; tt < 16; tt += 8) {
            int mi = (tt >> 2) * 16, ni = (tt & 3) * 16;
            v8f c = identC(mi, ni, lane);
            c = mmacc<false, false>(sL, NYY, mi, sM1, NXX, ni, NYY, c, lane);
            storeD(sP, NXX, mi, ni, c, lane);
        }
        __syncthreads();
    }
}

// ---------------------------------------------------------------------------
// Kernel 2: batch state recursion. 16 workgroups x 16 batch rows; per step:
//   xp = [x|u|d] @ [Wfx;Wfu;Wfd];  y = xp@Wfy;  x = xp + (ym - y)@G_t
// x-tile lives in LDS; G_t streamed from workspace (L2-resident, prefetched).
// ---------------------------------------------------------------------------
__global__ __launch_bounds__(128) void kf_batch(const float* __restrict__ Yp,
                                                const float* __restrict__ Up,
                                                const float* __restrict__ Dp,
                                                const float* __restrict__ Wfx,
                                                const float* __restrict__ Wfu,
                                                const float* __restrict__ Wfd,
                                                const float* __restrict__ Wfy,
                                                const float* __restrict__ G,
                                                float* __restrict__ out) {
    __shared__ float sW[(NXX + NUU + NDD) * NXX];  // stacked [88 x 64], 22.5 KB
    __shared__ float sWfy[NXX * NYY];              // 8 KB
    __shared__ float sx[16 * NXX];                 // state tile
    __shared__ float sxp[16 * NXX];                // predicted state tile
    __shared__ float su[16 * NUU];
    __shared__ float sd[16 * NDD];
    __shared__ float sym[16 * NYY];
    __shared__ float sin[16 * NYY];                // innovation

    const int tid = threadIdx.x, lane = tid & 31, w = tid >> 5;
    const int r0 = blockIdx.x * 16;

    for (int i = tid; i < NXX * NXX; i += 128) sW[i] = Wfx[i];
    for (int i = tid; i < NUU * NXX; i += 128) sW[NXX * NXX + i] = Wfu[i];
    for (int i = tid; i < NDD * NXX; i += 128) sW[(NXX + NUU) * NXX + i] = Wfd[i];
    for (int i = tid; i < NXX * NYY; i += 128) sWfy[i] = Wfy[i];
    for (int i = tid; i < 16 * NXX; i += 128) sx[i] = 0.0f;   // x0 = 0
    __syncthreads();

    for (int t = 0; t < T_STEPS; ++t) {
        const float* ut = Up + ((size_t)t * B_TOT + r0) * NUU;
        const float* dt = Dp + ((size_t)t * B_TOT + r0) * NDD;
        const float* yt = Yp + ((size_t)t * B_TOT + r0) * NYY;
        for (int i = tid; i < 16 * NUU; i += 128) su[i] = ut[i];
        for (int i = tid; i < 16 * NDD; i += 128) sd[i] = dt[i];
        for (int i = tid; i < 16 * NYY; i += 128) sym[i] = yt[i];
        const float* Gt = G + (size_t)t * (NYY * NXX);
        if (t + 1 < T_STEPS) {
            __builtin_prefetch(G + (size_t)(t + 1) * (NYY * NXX) + tid * 16, 0, 1);
        }
        __syncthreads();

        // stage 1: xp = x@Wfx + u@Wfu + d@Wfd  (wave w owns N-cols [16w,16w+16))
        {
            int n0 = w * 16;
            v8f c = zero8();
            c = mmacc<false, false>(sx, NXX, 0, sW, NXX, n0, NXX, c, lane);
            c = mmacc<false, false>(su, NUU, 0, sW + NXX * NXX, NXX, n0, NUU, c, lane);
            c = mmacc<false, false>(sd, NDD, 0, sW + (NXX + NUU) * NXX, NXX, n0, NDD, c, lane);
            storeD(sxp, NXX, 0, n0, c, lane);
        }
        __syncthreads();

        // stage 2: y = xp@Wfy ; innov = ym - y   (waves 0..1 cover N=32)
        if (w < 2) {
            int n0 = w * 16;
            v8f c = zero8();
            c = mmacc<false, false>(sxp, NXX, 0, sWfy, NYY, n0, NXX, c, lane);
            int h = lane >> 4, n = n0 + (lane & 15);
#pragma unroll
            for (int r = 0; r < 8; ++r) {
                int m = r + 8 * h;
                sin[m * NYY + n] = sym[m * NYY + n] - c[r];
            }
        }
        __syncthreads();

        // stage 3: x = xp + innov @ G_t   (G_t row-major [32 x 64], from L2)
        {
            int n0 = w * 16;
            v8f c = loadC(sxp, NXX, 0, n0, lane);
            c = mmacc<false, false>(sin, NYY, 0, Gt, NXX, n0, NYY, c, lane);
            storeD(sx, NXX, 0, n0, c, lane);
        }
        __syncthreads();
    }

    for (int i = tid; i < 16 * NXX; i += 128) {
        int r = i >> 6, cc = i & 63;
        out[(size_t)(r0 + r) * NXX + cc] = sx[i];
    }
}

extern "C" void kernel_launch(void* const* d_in, const int* in_sizes, int n_in,
                              void* d_out, int out_size, void* d_ws, size_t ws_size,
                              hipStream_t stream) {
    (void)in_sizes; (void)n_in; (void)out_size; (void)ws_size;
    const float* Yp  = (const float*)d_in[0];
    const float* Up  = (const float*)d_in[1];
    const float* Dp  = (const float*)d_in[2];
    const float* Wfx = (const float*)d_in[3];
    const float* Wfu = (const float*)d_in[4];
    const float* Wfd = (const float*)d_in[5];
    const float* Wfy = (const float*)d_in[6];
    float* G = (float*)d_ws;                 // 2048 * 32 * 64 * 4B = 16 MB
    float* out = (float*)d_out;              // [256 x 64] f32

    kf_gain_seq<<<1, 256, 0, stream>>>(Wfx, Wfy, G);
    kf_batch<<<16, 128, 0, stream>>>(Yp, Up, Dp, Wfx, Wfu, Wfd, Wfy, G, out);
}